// GraphEmbedder_5368709120132
// MI455X (gfx1250) — compile-verified
//
#include <hip/hip_runtime.h>

// ---------------------------------------------------------------------------
// GCN on MI455X (gfx1250, wave32).
// GEMM is tiny (4.2 GFLOP) -> done exactly with V_WMMA_F32_16X16X4_F32.
// Edge gather/scatter dominates (~4.4 GB of L2-resident traffic, 25.6MB
// feature matrix fits in the 192MB L2) -> wave-per-edge coalesced reads +
// f32 global atomics that resolve in L2.
// ---------------------------------------------------------------------------

typedef float v2f __attribute__((ext_vector_type(2)));
typedef float v8f __attribute__((ext_vector_type(8)));

#define GEMM_BLOCK 128   // 4 wave32s: wave w owns output columns [16w,16w+16)

static __host__ __device__ inline int cdiv(int a, int b) { return (a + b - 1) / b; }

// ---------------- degree / norm precompute ----------------

__global__ void k_set_one(float* __restrict__ deg, int N) {
    int i = blockIdx.x * blockDim.x + threadIdx.x;
    if (i < N) deg[i] = 1.0f;                      // self-loop weight
}

__global__ void k_deg_edges(const int* __restrict__ dst, const float* __restrict__ w,
                            float* __restrict__ deg, int E) {
    int e = blockIdx.x * blockDim.x + threadIdx.x;
    if (e < E) atomicAdd(&deg[dst[e]], w[e]);
}

__global__ void k_rsqrt_inplace(float* __restrict__ d, int N) {
    int i = blockIdx.x * blockDim.x + threadIdx.x;
    if (i < N) { float v = d[i]; d[i] = (v > 0.0f) ? rsqrtf(v) : 0.0f; }
}

__global__ void k_edge_norm(const int* __restrict__ src, const int* __restrict__ dst,
                            const float* __restrict__ w, const float* __restrict__ dis,
                            float* __restrict__ norm, int E) {
    int e = blockIdx.x * blockDim.x + threadIdx.x;
    if (e < E) norm[e] = dis[src[e]] * w[e] * dis[dst[e]];
}

// ---------------- WMMA f32 GEMM: Z[M,64] = H[M,K] @ W[K,64] ----------------
// One block = one 16-row tile. A tile (16xK) and full W (Kx64) staged in LDS.
// f32 16x16x4 fragment layouts per CDNA5 ISA 7.12.2:
//   A: lanes 0-15 = M rows; VGPR0 holds K=k0 (lo half) / k0+2 (hi half), VGPR1 = +1.
//   B: row-striped across lanes; VGPR0 = K row k0 (lo) / k0+2 (hi), VGPR1 = +1.
//   C/D: VGPR r -> M = r (lanes 0-15) or r+8 (lanes 16-31), N = lane&15.

__global__ void __launch_bounds__(GEMM_BLOCK)
k_gemm_wmma(const float* __restrict__ H, const float* __restrict__ Wm,
            float* __restrict__ Z, int M, int K) {
    __shared__ float ldsW[64 * 64];     // up to 16 KB
    __shared__ float ldsA[16 * 68];     // 16 rows, pitch K+4 (avoid bank conflicts)

    const int tid  = threadIdx.x;
    const int lane = tid & 31;
    const int wave = tid >> 5;
    const int row0 = blockIdx.x * 16;
    const int KP   = K + 4;

    for (int i = tid; i < K * 64; i += GEMM_BLOCK) ldsW[i] = Wm[i];
    for (int i = tid; i < 16 * K; i += GEMM_BLOCK) {
        int r = i / K, k = i - r * K;
        int grow = row0 + r;
        ldsA[r * KP + k] = (grow < M) ? H[grow * K + k] : 0.0f;
    }
    __syncthreads();

    const int ncol0 = wave << 4;
    const int half  = lane >> 4;    // 0: lanes 0-15, 1: lanes 16-31
    const int l15   = lane & 15;

    v8f acc = {0.f, 0.f, 0.f, 0.f, 0.f, 0.f, 0.f, 0.f};
    for (int k0 = 0; k0 < K; k0 += 4) {
        int kb = k0 + (half << 1);
        v2f a, b;
        a.x = ldsA[l15 * KP + kb];
        a.y = ldsA[l15 * KP + kb + 1];
        b.x = ldsW[kb * 64 + ncol0 + l15];
        b.y = ldsW[(kb + 1) * 64 + ncol0 + l15];
        acc = __builtin_amdgcn_wmma_f32_16x16x4_f32(
            /*neg_a=*/false, a, /*neg_b=*/false, b,
            /*c_mod=*/(short)0, acc, /*reuse_a=*/false, /*reuse_b=*/false);
    }

    const int col = ncol0 + l15;
#pragma unroll
    for (int r = 0; r < 8; ++r) {
        int row = row0 + r + (half << 3);
        if (row < M) Z[row * 64 + col] = acc[r];
    }
}

// ---------------- aggregate: init (self-loop + bias), edge scatter, relu ----

// X[i,j] = dis[i]^2 * Z[i,j] + b[j]   (self-loop norm = dis*1*dis, bias fused)
__global__ void k_init_next(const float* __restrict__ Z, const float* __restrict__ dis,
                            const float* __restrict__ bias, float* __restrict__ X, int NM) {
    int idx = blockIdx.x * blockDim.x + threadIdx.x;
    if (idx >= NM) return;
    int i = idx >> 6, j = idx & 63;
    float dn = dis[i];
    X[idx] = dn * dn * Z[idx] + bias[j];
}

// One wave32 per edge, each lane handles 2 columns (float2 coalesced read).
__global__ void k_scatter(const int* __restrict__ src, const int* __restrict__ dst,
                          const float* __restrict__ norm, const float* __restrict__ Z,
                          float* __restrict__ X, int E) {
    int idx = blockIdx.x * blockDim.x + threadIdx.x;
    int e = idx >> 5;
    if (e >= E) return;
    int j = (idx & 31) << 1;
    int s = src[e], d = dst[e];
    float nw = norm[e];
    float2 z = *(const float2*)(Z + (size_t)s * 64 + j);
    atomicAdd(X + (size_t)d * 64 + j,     nw * z.x);
    atomicAdd(X + (size_t)d * 64 + j + 1, nw * z.y);
}

__global__ void k_relu(float* __restrict__ X, int n) {
    int i = blockIdx.x * blockDim.x + threadIdx.x;
    if (i < n) X[i] = fmaxf(X[i], 0.0f);
}

// ---------------- segment-mean pool ----------------

__global__ void k_zero(float* __restrict__ p, int n) {
    int i = blockIdx.x * blockDim.x + threadIdx.x;
    if (i < n) p[i] = 0.0f;
}

__global__ void k_pool_sum(const int* __restrict__ batch, const float* __restrict__ X,
                           float* __restrict__ sums, float* __restrict__ cnts, int NM) {
    int idx = blockIdx.x * blockDim.x + threadIdx.x;
    if (idx >= NM) return;
    int i = idx >> 6, j = idx & 63;
    int g = batch[i];
    atomicAdd(&sums[g * 64 + j], X[idx]);
    if (j == 0) atomicAdd(&cnts[g], 1.0f);
}

__global__ void k_pool_final(const float* __restrict__ sums, const float* __restrict__ cnts,
                             float* __restrict__ out, int n) {
    int idx = blockIdx.x * blockDim.x + threadIdx.x;
    if (idx < n) out[idx] = sums[idx] / fmaxf(cnts[idx >> 6], 1.0f);
}

// ---------------------------------------------------------------------------

extern "C" void kernel_launch(void* const* d_in, const int* in_sizes, int n_in,
                              void* d_out, int out_size, void* d_ws, size_t ws_size,
                              hipStream_t stream) {
    const float* x     = (const float*)d_in[0];   // [N,32]
    const int*   ei    = (const int*)  d_in[1];   // [2,E]
    const float* ew    = (const float*)d_in[2];   // [E]
    const int*   batch = (const int*)  d_in[3];   // [N]
    const float* W_in  = (const float*)d_in[4];   // [32,64]
    const float* b_in  = (const float*)d_in[5];   // [64]
    const float* W_h   = (const float*)d_in[6];   // [3,64,64]
    const float* b_h   = (const float*)d_in[7];   // [3,64]
    const float* W_out = (const float*)d_in[8];   // [64,64]
    const float* b_out = (const float*)d_in[9];   // [64]

    const int E = in_sizes[2];
    const int N = in_sizes[3];
    const int* src = ei;
    const int* dst = ei + E;
    const int NM = N * 64;
    const int NUM_GRAPHS = 256;

    // workspace layout (floats): dis[N] | norm[E] | bufX[N*64] | bufY[N*64] | sums[256*64] | cnts[256]
    float* ws   = (float*)d_ws;
    float* dis  = ws;
    float* norm = dis + N;
    float* bufX = norm + E;
    float* bufY = bufX + (size_t)NM;
    float* sums = bufY + (size_t)NM;
    float* cnts = sums + NUM_GRAPHS * 64;

    const int T = 256;

    // --- degree + symmetric normalization (deg computed in-place in `dis`) ---
    k_set_one     <<<cdiv(N, T), T, 0, stream>>>(dis, N);
    k_deg_edges   <<<cdiv(E, T), T, 0, stream>>>(dst, ew, dis, E);
    k_rsqrt_inplace<<<cdiv(N, T), T, 0, stream>>>(dis, N);
    k_edge_norm   <<<cdiv(E, T), T, 0, stream>>>(src, dst, ew, dis, norm, E);

    const int gemm_grid    = cdiv(N, 16);
    const int scatter_grid = cdiv(E * 32, T);
    const int nm_grid      = cdiv(NM, T);

    // --- layer 1: x[N,32] @ W_in ---
    k_gemm_wmma <<<gemm_grid, GEMM_BLOCK, 0, stream>>>(x, W_in, bufY, N, 32);
    k_init_next <<<nm_grid, T, 0, stream>>>(bufY, dis, b_in, bufX, NM);
    k_scatter   <<<scatter_grid, T, 0, stream>>>(src, dst, norm, bufY, bufX, E);
    k_relu      <<<nm_grid, T, 0, stream>>>(bufX, NM);

    // --- hidden layers ---
    for (int l = 0; l < 3; ++l) {
        k_gemm_wmma <<<gemm_grid, GEMM_BLOCK, 0, stream>>>(bufX, W_h + l * 64 * 64, bufY, N, 64);
        k_init_next <<<nm_grid, T, 0, stream>>>(bufY, dis, b_h + l * 64, bufX, NM);
        k_scatter   <<<scatter_grid, T, 0, stream>>>(src, dst, norm, bufY, bufX, E);
        k_relu      <<<nm_grid, T, 0, stream>>>(bufX, NM);
    }

    // --- output layer (no relu) ---
    k_gemm_wmma <<<gemm_grid, GEMM_BLOCK, 0, stream>>>(bufX, W_out, bufY, N, 64);
    k_init_next <<<nm_grid, T, 0, stream>>>(bufY, dis, b_out, bufX, NM);
    k_scatter   <<<scatter_grid, T, 0, stream>>>(src, dst, norm, bufY, bufX, E);

    // --- global mean pool -> d_out [16,16,64] (contiguous row-major) ---
    k_zero      <<<cdiv(NUM_GRAPHS * 64 + NUM_GRAPHS, T), T, 0, stream>>>(sums, NUM_GRAPHS * 64 + NUM_GRAPHS);
    k_pool_sum  <<<nm_grid, T, 0, stream>>>(batch, bufX, sums, cnts, NM);
    k_pool_final<<<cdiv(NUM_GRAPHS * 64, T), T, 0, stream>>>(sums, cnts, (float*)d_out, NUM_GRAPHS * 64);
}